// ShapeletLayer_43611097923753
// MI455X (gfx1250) — compile-verified
//
#include <hip/hip_runtime.h>

typedef float v2f __attribute__((ext_vector_type(2)));
typedef float v8f __attribute__((ext_vector_type(8)));

constexpr int N_ = 64, C_ = 4, D_ = 2048, K_ = 64, O_ = 256;
constexpr int L_ = D_ - K_ + 1;   // 1985
constexpr int LDS_WPAD = 65;      // weight row stride: conflict-free B-frag reads
constexpr int TILE_PAD = 19;      // out-tile row stride: conflict-free on write & read

// One workgroup: 8 waves (wave32). Each wave -> one 16(l) x 16(o) f32 tile.
// Waves share the l-tile's x window (LDS) and a 128-row weight slab (LDS).
// Epilogue transposes each wave's tile through LDS so output stores are
// 64B-dense per row segment (kernel is store-bandwidth-bound: ~520MB out).
__global__ __launch_bounds__(256) void shapelet_wmma_kernel(
    const float* __restrict__ x,      // [N, C, D]
    const float* __restrict__ w,      // [O, C, K]
    float* __restrict__ out)          // [N, C, O, L]
{
    __shared__ float xs[80];                    // x[l_base .. l_base+78] (+pad)
    __shared__ float xn[16];                    // x_norm per l in tile
    __shared__ float yn[128];                   // y_norm per o in slab
    __shared__ float ws[128 * LDS_WPAD];        // weight slab [o_local][k], padded
    __shared__ float tile[8 * 16 * TILE_PAD];   // per-wave 16x16 result tiles

    const int t      = threadIdx.x;
    const int l_base = blockIdx.x * 16;         // 125 tiles cover L=1985
    const int o_base = blockIdx.y * 128;        // 2 slabs cover O=256
    const int nc     = blockIdx.z;              // n*C + c
    const int c      = nc & (C_ - 1);

    // Stage x window: 16 l-values + K-1 reach = 79 floats
    if (t < 80) {
        const int d = l_base + t;
        xs[t] = (d < D_) ? x[(size_t)nc * D_ + d] : 0.0f;
    }
    // Stage weight slab: 128 o-rows x 64 k  (coalesced: consecutive lanes -> consecutive k)
    {
        const float* wsrc = w + (size_t)o_base * (C_ * K_) + (size_t)c * K_;
        #pragma unroll
        for (int i = 0; i < 32; ++i) {
            const int idx = t + 256 * i;
            const int ol  = idx >> 6;
            const int k   = idx & 63;
            ws[ol * LDS_WPAD + k] = wsrc[(size_t)ol * (C_ * K_) + k];
        }
    }
    __syncthreads();

    // y_norm[o_local] = sum_k w^2   (row-per-thread; pad keeps banks distinct)
    if (t < 128) {
        float s = 0.0f;
        #pragma unroll 8
        for (int k = 0; k < K_; ++k) { const float v = ws[t * LDS_WPAD + k]; s += v * v; }
        yn[t] = s;
    }
    // x_norm[m] = sum_k x[l_base+m+k]^2
    if (t < 16) {
        float s = 0.0f;
        #pragma unroll 8
        for (int k = 0; k < K_; ++k) { const float v = xs[t + k]; s += v * v; }
        xn[t] = s;
    }
    __syncthreads();

    const int lane = t & 31;
    const int wv   = t >> 5;                  // wave id 0..7 -> o sub-tile
    const int ncol = lane & 15;               // A row m / B column n
    const int koff = (lane >> 4) << 1;        // K-pair select: 0 (lanes 0-15) / 2 (lanes 16-31)
    const int wrow = wv * 16 + ncol;          // o_local for this lane's B column

    // C/D accumulator: 16x16 f32, 8 VGPRs per lane (ISA 7.12.2 layout)
    v8f acc = {};

    // K=64 reduction as 16 x V_WMMA_F32_16X16X4_F32 (full f32 precision, 2048 FLOP each)
    #pragma unroll
    for (int kk = 0; kk < 16; ++kk) {
        const int kidx = kk * 4 + koff;
        v2f a, b;
        a.x = xs[ncol + kidx];                       // A[m][k]   = x[l_base+m+k]
        a.y = xs[ncol + kidx + 1];                   // A[m][k+1]
        b.x = ws[wrow * LDS_WPAD + kidx];            // B[k][n]   = w[o_base+wrow][k]
        b.y = ws[wrow * LDS_WPAD + kidx + 1];        // B[k+1][n]
        acc = __builtin_amdgcn_wmma_f32_16x16x4_f32(
                  /*neg_a=*/false, a, /*neg_b=*/false, b,
                  /*c_mod=*/(short)0, acc, /*reuse_a=*/false, /*reuse_b=*/false);
    }

    // Apply epilogue in-register, park finished tile in LDS.
    // acc[j] is element (m = moff+j [l-dim], n = ncol [o-dim]) of the tile.
    {
        const float ynv  = yn[wrow];
        const int   moff = (lane >> 4) << 3;
        float* tw = tile + wv * (16 * TILE_PAD) + ncol * TILE_PAD + moff;
        #pragma unroll
        for (int j = 0; j < 8; ++j) {
            float v = xn[moff + j] + ynv - 2.0f * acc[j];
            tw[j] = fmaxf(v, 0.0f) * (1.0f / (float)K_);
        }
    }
    __syncthreads();

    // Coalesced store: per instruction, lanes 0-15 write one full 64B row
    // segment (fixed o, l_base..l_base+15), lanes 16-31 the next row.
    {
        const int    col  = lane & 15;            // l within tile
        const int    row0 = lane >> 4;            // o parity within pair
        const int    l    = l_base + col;
        const float* tr   = tile + wv * (16 * TILE_PAD) + col;
        const bool   lok  = (l < L_);
        #pragma unroll
        for (int i = 0; i < 8; ++i) {
            const int row = row0 + 2 * i;         // o within wave tile
            const float v = tr[row * TILE_PAD];
            if (lok) {
                const size_t oidx = ((size_t)nc * O_ + (size_t)(o_base + wv * 16 + row)) * (size_t)L_ + l;
                out[oidx] = v;
            }
        }
    }
}

extern "C" void kernel_launch(void* const* d_in, const int* in_sizes, int n_in,
                              void* d_out, int out_size, void* d_ws, size_t ws_size,
                              hipStream_t stream) {
    (void)in_sizes; (void)n_in; (void)out_size; (void)d_ws; (void)ws_size;
    const float* x   = (const float*)d_in[0];   // [N, C, D] f32
    const float* w   = (const float*)d_in[1];   // [O, C, K] f32
    float*       out = (float*)d_out;           // [N, C, O, L] f32

    dim3 grid((L_ + 15) / 16,   // 125 l-tiles
              O_ / 128,         // 2 o-slabs
              N_ * C_);         // 256 (n,c) pairs
    shapelet_wmma_kernel<<<grid, dim3(256), 0, stream>>>(x, w, out);
}